// Attention_26903675142688
// MI455X (gfx1250) — compile-verified
//
#include <hip/hip_runtime.h>

// ---------------------------------------------------------------------------
// Masked single-query attention, fp32, for MI455X (gfx1250).
//   query [B,D], key/value [B,T,D], mask [B,T] -> (context [B,D], attn [B,T])
// B=64, T=4096, D=512. Memory-bound (~1.07 GB / 23.3 TB/s ~= 46 us).
// Energy GEMV runs on V_WMMA_F32_16X16X4_F32 (exact fp32 matrix path).
// ---------------------------------------------------------------------------

typedef float v2f __attribute__((ext_vector_type(2)));
typedef float v8f __attribute__((ext_vector_type(8)));

constexpr int   B_    = 64;
constexpr int   T_    = 4096;
constexpr int   D_    = 512;
constexpr int   S_    = 8;           // segments per batch row (split-T)
constexpr int   TSEG  = T_ / S_;     // 512
constexpr float SCALE = 0.04419417382415922f;   // 1/sqrt(512)
constexpr float BIGN  = 1000000000.0f;

// ---------------------------------------------------------------------------
// Kernel A: per (batch, segment): energies via WMMA, segment softmax,
// partial context. 256 threads = 8 wave32s.
// ---------------------------------------------------------------------------
__global__ __launch_bounds__(256)
void attn_partial(const float* __restrict__ query,
                  const float* __restrict__ key,
                  const float* __restrict__ value,
                  const int*   __restrict__ mask,
                  float* __restrict__ out_attn,   // receives p = exp(e - m_seg)
                  float* __restrict__ wsM,        // [B*S] segment max
                  float* __restrict__ wsS,        // [B*S] segment sum of exp
                  float* __restrict__ wsPC)       // [B*S, D] partial context
{
    __shared__ float q_lds[D_];
    __shared__ float e_lds[TSEG];
    __shared__ float red[256];

    const int tid   = threadIdx.x;
    const int b     = blockIdx.x / S_;
    const int seg   = blockIdx.x % S_;
    const int tbase = seg * TSEG;

    q_lds[tid]       = query[b * D_ + tid];
    q_lds[tid + 256] = query[b * D_ + tid + 256];
    __syncthreads();

    // ---- Phase 1: energy[t] = K[t,:].q via V_WMMA_F32_16X16X4_F32 ----------
    // A (16x4 f32) layout: lanes 0-15 -> M=lane, VGPR0/1 = k0,k1
    //                      lanes 16-31 -> M=lane-16, VGPR0/1 = k2,k3
    // B (4x16) = q-chunk replicated across the 16 columns -> every column of
    // D holds the energies; extract column N=0 from lanes 0 (M=0..7) and
    // 16 (M=8..15), one value per accumulator VGPR.
    const int lane = tid & 31;
    const int wave = tid >> 5;
    const int half = lane >> 4;   // 0: holds k0,k1   1: holds k2,k3
    const int row  = lane & 15;   // A-matrix row M

    for (int tile = 0; tile < TSEG / (16 * 8); ++tile) {   // 4 tiles / wave
        const int tl = (wave * 4 + tile) * 16;
        const float* kp = key + (long)(b * T_ + tbase + tl + row) * D_ + 2 * half;
        const float* qp = q_lds + 2 * half;
        v8f c = {0.f, 0.f, 0.f, 0.f, 0.f, 0.f, 0.f, 0.f};
        #pragma unroll 4
        for (int d0 = 0; d0 < D_; d0 += 4) {
            v2f a  = *(const v2f*)(kp + d0);
            v2f bq = *(const v2f*)(qp + d0);
            c = __builtin_amdgcn_wmma_f32_16x16x4_f32(false, a, false, bq,
                                                      (short)0, c, false, false);
        }
        if (row == 0) {           // lanes 0 and 16 hold column N=0
            const int base = tl + half * 8;
            #pragma unroll
            for (int r = 0; r < 8; ++r) e_lds[base + r] = c[r] * SCALE;
        }
    }
    __syncthreads();

    // ---- Phase 2: mask + segment-local softmax -----------------------------
    const int t0 = tid, t1 = tid + 256;
    float e0 = e_lds[t0] + (float)(mask[(long)b * T_ + tbase + t0] - 1) * BIGN;
    float e1 = e_lds[t1] + (float)(mask[(long)b * T_ + tbase + t1] - 1) * BIGN;

    red[tid] = fmaxf(e0, e1);
    __syncthreads();
    for (int off = 128; off > 0; off >>= 1) {
        if (tid < off) red[tid] = fmaxf(red[tid], red[tid + off]);
        __syncthreads();
    }
    const float m = red[0];
    __syncthreads();

    const float p0 = __expf(e0 - m);
    const float p1 = __expf(e1 - m);
    e_lds[t0] = p0;
    e_lds[t1] = p1;
    out_attn[(long)b * T_ + tbase + t0] = p0;   // rescaled by kernel B
    out_attn[(long)b * T_ + tbase + t1] = p1;

    red[tid] = p0 + p1;
    __syncthreads();
    for (int off = 128; off > 0; off >>= 1) {
        if (tid < off) red[tid] += red[tid + off];
        __syncthreads();
    }
    if (tid == 0) { wsM[blockIdx.x] = m; wsS[blockIdx.x] = red[0]; }

    // ---- Phase 3: partial context  pc[d] = sum_t p[t] * V[t,d] -------------
    // Coalesced float2 stream of V (2 KB per block per step).
    const float* vp = value + (long)(b * T_ + tbase) * D_ + 2 * tid;
    float ax = 0.f, ay = 0.f;
    #pragma unroll 4
    for (int t = 0; t < TSEG; ++t) {
        const float p  = e_lds[t];
        const v2f   vv = *(const v2f*)(vp + (long)t * D_);
        ax = fmaf(p, vv.x, ax);
        ay = fmaf(p, vv.y, ay);
    }
    float* pc = wsPC + (long)blockIdx.x * D_ + 2 * tid;
    pc[0] = ax;
    pc[1] = ay;
}

// ---------------------------------------------------------------------------
// Kernel B: merge the S segments per batch (log-sum-exp combine), write final
// context and rescale attn in place.
// ---------------------------------------------------------------------------
__global__ __launch_bounds__(256)
void attn_combine(const float* __restrict__ wsM,
                  const float* __restrict__ wsS,
                  const float* __restrict__ wsPC,
                  float* __restrict__ out_ctx,
                  float* __restrict__ out_attn)
{
    const int b   = blockIdx.x;
    const int tid = threadIdx.x;

    float ms[S_];
    #pragma unroll
    for (int s = 0; s < S_; ++s) ms[s] = wsM[b * S_ + s];
    float gm = ms[0];
    #pragma unroll
    for (int s = 1; s < S_; ++s) gm = fmaxf(gm, ms[s]);

    float sc[S_];
    float Z = 0.f;
    #pragma unroll
    for (int s = 0; s < S_; ++s) {
        sc[s] = __expf(ms[s] - gm);
        Z += wsS[b * S_ + s] * sc[s];
    }
    const float inv = 1.0f / Z;

    // final context
    for (int d = tid; d < D_; d += 256) {
        float acc = 0.f;
        #pragma unroll
        for (int s = 0; s < S_; ++s)
            acc = fmaf(wsPC[(long)(b * S_ + s) * D_ + d], sc[s], acc);
        out_ctx[b * D_ + d] = acc * inv;
    }

    // final attn = p * exp(m_seg - M) / Z
    for (int t = tid; t < T_; t += 256) {
        out_attn[(long)b * T_ + t] *= sc[t >> 9] * inv;   // TSEG = 512
    }
}

// ---------------------------------------------------------------------------
extern "C" void kernel_launch(void* const* d_in, const int* in_sizes, int n_in,
                              void* d_out, int out_size, void* d_ws, size_t ws_size,
                              hipStream_t stream)
{
    (void)in_sizes; (void)n_in; (void)out_size; (void)ws_size;

    const float* query = (const float*)d_in[0];
    const float* key   = (const float*)d_in[1];
    const float* value = (const float*)d_in[2];
    const int*   mask  = (const int*)d_in[3];

    float* out_ctx  = (float*)d_out;             // [B, D]
    float* out_attn = out_ctx + B_ * D_;         // [B, T]

    float* ws   = (float*)d_ws;
    float* wsM  = ws;                            // [B*S]
    float* wsS  = ws + B_ * S_;                  // [B*S]
    float* wsPC = ws + 2 * B_ * S_;              // [B*S, D]  (~1 MB total)

    attn_partial<<<B_ * S_, 256, 0, stream>>>(query, key, value, mask,
                                              out_attn, wsM, wsS, wsPC);
    attn_combine<<<B_, 256, 0, stream>>>(wsM, wsS, wsPC, out_ctx, out_attn);
}